// FakeQuant_pow2_88991722373171
// MI455X (gfx1250) — compile-verified
//
#include <hip/hip_runtime.h>
#include <math.h>

// 128-bit vector type usable with __builtin_nontemporal_load/store
typedef __attribute__((ext_vector_type(4))) float v4f;

// Monotonic float -> uint key: unsigned order of key == float order of value.
static __device__ __forceinline__ unsigned int fkey(float f) {
    unsigned int u = __float_as_uint(f);
    return (u & 0x80000000u) ? ~u : (u ^ 0x80000000u);
}
static __device__ __forceinline__ float fdec(unsigned int k) {
    unsigned int u = (k & 0x80000000u) ? (k ^ 0x80000000u) : ~k;
    return __uint_as_float(u);
}

// ---------------------------------------------------------------------------
// Kernel 0: seed workspace identities (harness poisons d_ws with 0xAA).
// ws[0] = running MIN key (identity 0xFFFFFFFF), ws[1] = MAX key (identity 0)
// ---------------------------------------------------------------------------
__global__ void fq_init_kernel(unsigned int* __restrict__ ws) {
    ws[0] = 0xFFFFFFFFu;
    ws[1] = 0x00000000u;
}

// ---------------------------------------------------------------------------
// Kernel 1: global min/max reduction using the CDNA5 async-to-LDS engine.
// Each block walks 4 KB chunks (256 lanes x 16 B). Double-buffered:
//   issue global_load_async_to_lds_b128 for chunk k+1  (ASYNCcnt++)
//   s_wait_asynccnt 0x1  -> oldest async op (chunk k) has landed in LDS
//   ds_load current stage, reduce.
// s_wait_dscnt 0x0 before each issue guarantees the prior ds_load finished
// reading the slot the async engine is about to overwrite.
// ---------------------------------------------------------------------------
__global__ void __launch_bounds__(256)
fq_minmax_kernel(const float* __restrict__ x, unsigned int* __restrict__ ws,
                 unsigned int nchunk, unsigned int nvec, unsigned int n) {
    const v4f* __restrict__ xv = (const v4f*)x;
    const unsigned int tid = threadIdx.x;

    float mn =  __builtin_inff();
    float mx = -__builtin_inff();

#if defined(__AMDGCN__)
    __shared__ v4f stage[2][256];
    // addrspace(3) lives in the low 32 bits of the generic VA (flat aperture)
    const unsigned int lds0 = (unsigned int)(size_t)&stage[0][tid];
    const unsigned int lds1 = (unsigned int)(size_t)&stage[1][tid];

    unsigned int c        = blockIdx.x;
    const unsigned int cs = gridDim.x;

    if (c < nchunk) {  // prologue: stage chunk c into buffer 0
        unsigned long long g = (unsigned long long)(size_t)(xv + ((size_t)c * 256u + tid));
        asm volatile("global_load_async_to_lds_b128 %0, %1, off"
                     :: "v"(lds0), "v"(g) : "memory");
    }
    unsigned int buf = 0;
    for (; c < nchunk; c += cs) {
        const unsigned int cn = c + cs;
        asm volatile("s_wait_dscnt 0x0" ::: "memory");  // slot free for reuse
        if (cn < nchunk) {
            unsigned long long g =
                (unsigned long long)(size_t)(xv + ((size_t)cn * 256u + tid));
            const unsigned int l = buf ? lds0 : lds1;
            asm volatile("global_load_async_to_lds_b128 %0, %1, off"
                         :: "v"(l), "v"(g) : "memory");
            asm volatile("s_wait_asynccnt 0x1" ::: "memory");  // chunk c landed
        } else {
            asm volatile("s_wait_asynccnt 0x0" ::: "memory");  // drain
        }
        v4f v = stage[buf][tid];                                // ds_load_b128
        mn = fminf(mn, fminf(fminf(v[0], v[1]), fminf(v[2], v[3])));
        mx = fmaxf(mx, fmaxf(fmaxf(v[0], v[1]), fmaxf(v[2], v[3])));
        buf ^= 1u;
    }
#else
    for (unsigned int j = blockIdx.x * 256u + tid; j < nchunk * 256u;
         j += gridDim.x * 256u) {
        v4f v = xv[j];
        mn = fminf(mn, fminf(fminf(v[0], v[1]), fminf(v[2], v[3])));
        mx = fmaxf(mx, fmaxf(fmaxf(v[0], v[1]), fmaxf(v[2], v[3])));
    }
#endif

    // Remainder vec4 chunks (< 256) and scalar tail (< 4): block 0 only.
    if (blockIdx.x == 0) {
        for (unsigned int j = nchunk * 256u + tid; j < nvec; j += 256u) {
            v4f v = xv[j];
            mn = fminf(mn, fminf(fminf(v[0], v[1]), fminf(v[2], v[3])));
            mx = fmaxf(mx, fmaxf(fmaxf(v[0], v[1]), fmaxf(v[2], v[3])));
        }
        if (tid == 0) {
            for (unsigned int k = nvec * 4u; k < n; ++k) {
                mn = fminf(mn, x[k]);
                mx = fmaxf(mx, x[k]);
            }
        }
    }

    // wave32 reduction tree
    #pragma unroll
    for (int off = 16; off > 0; off >>= 1) {
        mn = fminf(mn, __shfl_xor(mn, off, 32));
        mx = fmaxf(mx, __shfl_xor(mx, off, 32));
    }
    if ((tid & 31u) == 0u) {
        atomicMin(&ws[0], fkey(mn));   // global_atomic_min_u32
        atomicMax(&ws[1], fkey(mx));   // global_atomic_max_u32
    }
}

// ---------------------------------------------------------------------------
// Kernel 2: apply fake-quant, traversing in REVERSE so we start on the data
// pass 1 left resident in the 192 MB L2. Output stored non-temporally so the
// write stream does not evict input lines we want to re-hit.
// ---------------------------------------------------------------------------
__global__ void __launch_bounds__(256)
fq_apply_kernel(const float* __restrict__ x, float* __restrict__ out,
                const unsigned int* __restrict__ ws,
                unsigned int nvec, unsigned int n) {
    const float mn = fdec(ws[0]);                 // uniform -> s_load path
    const float mx = fdec(ws[1]);
    const float scale = 255.0f / (mx - mn);
    const float inv   = 1.0f / scale;             // hoisted real division

    const v4f* __restrict__ xv = (const v4f*)x;
    v4f* __restrict__ ov       = (v4f*)out;

    unsigned int i            = blockIdx.x * blockDim.x + threadIdx.x;
    const unsigned int stride = gridDim.x * blockDim.x;

    for (; i < nvec; i += stride) {
        const unsigned int idx = nvec - 1u - i;   // reversed: L2-hot tail first
        if (i + 8u * stride < nvec)
            __builtin_prefetch(xv + (nvec - 1u - (i + 8u * stride)), 0, 1);
        v4f v = xv[idx];                          // global_load_b128 (RT)
        v4f r;
        #pragma unroll
        for (int cpt = 0; cpt < 4; ++cpt) {
            float t = (v[cpt] - mn) * scale;
            r[cpt] = fmaf(rintf(t), inv, mn);     // v_rndne_f32 (half-to-even)
        }
        __builtin_nontemporal_store(r, ov + idx); // TH=NT global_store_b128
    }

    if (blockIdx.x == 0 && threadIdx.x == 0) {
        for (unsigned int k = nvec * 4u; k < n; ++k) {
            float t = (x[k] - mn) * scale;
            out[k] = fmaf(rintf(t), inv, mn);
        }
    }
}

// ---------------------------------------------------------------------------
// Host-side launcher
// ---------------------------------------------------------------------------
extern "C" void kernel_launch(void* const* d_in, const int* in_sizes, int n_in,
                              void* d_out, int out_size, void* d_ws, size_t ws_size,
                              hipStream_t stream) {
    const float* x   = (const float*)d_in[0];
    float* out       = (float*)d_out;
    unsigned int* ws = (unsigned int*)d_ws;

    const unsigned int n      = (unsigned int)in_sizes[0];
    const unsigned int nvec   = n >> 2;          // float4 chunks
    const unsigned int nchunk = nvec / 256u;     // 4 KB block-chunks

    const int BLOCK = 256;
    unsigned int want_r = nchunk > 0u ? nchunk : 1u;
    int grid_r = (int)(want_r < 3072u ? want_r : 3072u);

    unsigned int want_a = (nvec + BLOCK - 1) / BLOCK;
    if (want_a == 0u) want_a = 1u;
    int grid_a = (int)(want_a < 3072u ? want_a : 3072u);

    fq_init_kernel<<<1, 1, 0, stream>>>(ws);
    fq_minmax_kernel<<<grid_r, BLOCK, 0, stream>>>(x, ws, nchunk, nvec, n);
    fq_apply_kernel<<<grid_a, BLOCK, 0, stream>>>(x, out, ws, nvec, n);
}